// MaxoutDynamic_55181739819231
// MI455X (gfx1250) — compile-verified
//
#include <hip/hip_runtime.h>

#define FEAT   4096
#define NDROP  2048      // rank (0-based, ascending) of the threshold element
#define TPB    256
#define NCHUNK 4         // 4 chunks of float4 per thread -> 16 elements
#define NWAVE  8         // 256 threads / wave32

typedef float    v4f __attribute__((ext_vector_type(4)));
typedef unsigned v4u __attribute__((ext_vector_type(4)));

__device__ __forceinline__ unsigned f2key(float f) {
  // order-preserving float -> uint map (ascending)
  unsigned u = __float_as_uint(f);
  return u ^ (unsigned)(((int)u >> 31) | 0x80000000);
}

__global__ __launch_bounds__(TPB) void MaxoutDynamic_radixselect_kernel(
    const float* __restrict__ feat, float* __restrict__ out) {
  __shared__ __attribute__((aligned(16))) float rowbuf[FEAT];  // 16 KB staged row
  __shared__ int      hist[NWAVE * 256];                       // 8 KB per-wave histograms
  __shared__ int      wsum[NWAVE];
  __shared__ unsigned sel[2];                                  // {selected digit, new rank}

  const int      row  = blockIdx.x;
  const int      t    = threadIdx.x;
  const unsigned lane = (unsigned)t & 31u;
  const unsigned wv   = (unsigned)t >> 5;

  const float* rowp = feat + (size_t)row * FEAT;

  // LDS byte offset of rowbuf: low 32 bits of the flat LDS pointer are the
  // wave-relative LDS address (ISA 10.2 aperture mapping).
  unsigned lds_row = (unsigned)(unsigned long long)(const void*)&rowbuf[0];
  unsigned long long gbase = (unsigned long long)rowp;

  // ---- CDNA5 async DMA: global row -> LDS (4 x 256 lanes x 16B = 16 KB) ----
#pragma unroll
  for (int j = 0; j < 4; ++j) {
    unsigned byteoff = (unsigned)((j * TPB + t) * 16);
    unsigned ldsaddr = lds_row + byteoff;
    asm volatile("global_load_async_to_lds_b128 %0, %1, %2"
                 :
                 : "v"(ldsaddr), "v"(byteoff), "s"(gbase)
                 : "memory");
  }
  asm volatile("s_wait_asynccnt 0" ::: "memory");
  __syncthreads();

  // ---- pull 4 x float4 per thread into VGPRs (ds_load_b128) ----
  v4f x[NCHUNK];
  v4u key[NCHUNK];
#pragma unroll
  for (int g = 0; g < NCHUNK; ++g) {
    x[g] = *(const v4f*)&rowbuf[g * (TPB * 4) + t * 4];
#pragma unroll
    for (int c = 0; c < 4; ++c) key[g][c] = f2key(x[g][c]);
  }

  // ---- exact radix select of the rank-NDROP key (MSB-first, 8-bit digits) ----
  unsigned prefix = 0u;
  int      r      = NDROP;
#pragma unroll 1
  for (int d = 3; d >= 0; --d) {
    // zero all per-wave histograms
#pragma unroll
    for (int w = 0; w < NWAVE; ++w) hist[w * 256 + t] = 0;
    __syncthreads();

    const unsigned shift  = 8u * (unsigned)d;
    const unsigned himask = (d == 3) ? 0u : (0xFFFFFFFFu << (8u * (unsigned)(d + 1)));
    int* myhist = &hist[wv * 256];
#pragma unroll
    for (int g = 0; g < NCHUNK; ++g) {
#pragma unroll
      for (int c = 0; c < 4; ++c) {
        if (((key[g][c] ^ prefix) & himask) == 0u) {
          atomicAdd(&myhist[(key[g][c] >> shift) & 255u], 1);
        }
      }
    }
    __syncthreads();

    // reduce 8 wave-histograms, then block-wide exclusive scan over 256 bins
    int cnt = 0;
#pragma unroll
    for (int w = 0; w < NWAVE; ++w) cnt += hist[w * 256 + t];

    int inc = cnt;                       // wave32 inclusive scan
#pragma unroll
    for (int o = 1; o < 32; o <<= 1) {
      int nv = __shfl_up(inc, (unsigned)o, 32);
      if ((int)lane >= o) inc += nv;
    }
    if (lane == 31u) wsum[wv] = inc;
    __syncthreads();
    if (t == 0) {
      int s = 0;
#pragma unroll
      for (int w = 0; w < NWAVE; ++w) { int v = wsum[w]; wsum[w] = s; s += v; }
    }
    __syncthreads();

    int excl = inc - cnt + wsum[wv];
    if (cnt > 0 && excl <= r && r < excl + cnt) {  // exactly one thread matches
      sel[0] = (unsigned)t;
      sel[1] = (unsigned)(r - excl);
    }
    __syncthreads();
    prefix |= sel[0] << shift;
    r = (int)sel[1];
  }

  // ---- masked scale + vectorized non-temporal HBM write (b128, TH=NT) ----
  const unsigned T = prefix;             // exact key of the rank-2048 element
  float* orow = out + (size_t)row * FEAT;
#pragma unroll
  for (int g = 0; g < NCHUNK; ++g) {
    v4f res;
#pragma unroll
    for (int c = 0; c < 4; ++c) {
      res[c] = (key[g][c] >= T) ? x[g][c] * 2.0f : 0.0f;
    }
    __builtin_nontemporal_store(res, (v4f*)&orow[g * (TPB * 4) + t * 4]);
  }
}

extern "C" void kernel_launch(void* const* d_in, const int* in_sizes, int n_in,
                              void* d_out, int out_size, void* d_ws, size_t ws_size,
                              hipStream_t stream) {
  (void)n_in; (void)d_ws; (void)ws_size; (void)out_size;
  const float* feat = (const float*)d_in[0];
  float*       out  = (float*)d_out;
  const int rows = in_sizes[0] / FEAT;   // 16384
  MaxoutDynamic_radixselect_kernel<<<rows, TPB, 0, stream>>>(feat, out);
}